// PostProcess_10728828306054
// MI455X (gfx1250) — compile-verified
//
#include <hip/hip_runtime.h>
#include <hip/hip_bf16.h>
#include <math.h>

// Problem constants (match reference)
#define BB 64
#define QQ 900
#define CC 365
#define BQ (BB * QQ)
#define HUMAN_CONF 0.7f
#define THRESH 0.25f
#define IOU_THRESH 0.5f

#define QPAD 1024      // power-of-2 pad for bitonic sort of 900 entries
#define KPAD 368       // 365 padded to multiple of 4
#define MTILES 3600    // 57600/16
#define WAVES_PER_WG 8
#define NT_PER_WG 2    // N-tiles per workgroup (A reuse)
#define NGROUPS 12     // ceil(23/2) -> grid.y

typedef __attribute__((ext_vector_type(2))) float v2f;
typedef __attribute__((ext_vector_type(8))) float v8f;
typedef __attribute__((ext_vector_type(4))) unsigned u32x4;
typedef __attribute__((ext_vector_type(8))) int i32x8;
typedef __attribute__((ext_vector_type(4))) int i32x4;

#if defined(__has_builtin)
#if __has_builtin(__builtin_amdgcn_tensor_load_to_lds) && __has_builtin(__builtin_amdgcn_s_wait_tensorcnt)
#define USE_TDM 1
#endif
#endif

// ---------------------------------------------------------------------------
// Kernel 1: pointwise prep — human softmax / labels / valid, boxes cxcywh->xyxy
// scaled, int-cast boxes to output.
// ---------------------------------------------------------------------------
__global__ void prep_kernel(const float* __restrict__ out_human,
                            const float* __restrict__ out_bbox,
                            const int* __restrict__ p_h,
                            const int* __restrict__ p_w,
                            float* __restrict__ ws_boxes,   // [BQ,4] xyxy scaled
                            float* __restrict__ ws_hscore,  // [BQ]
                            int*   __restrict__ ws_valid,   // [BQ]
                            float* __restrict__ out_boxes)  // [BQ,4] truncated-int as float
{
    int i = blockIdx.x * blockDim.x + threadIdx.x;
    if (i >= BQ) return;

    float h0 = out_human[2 * i + 0];
    float h1 = out_human[2 * i + 1];
    float m  = fmaxf(h0, h1);
    float e0 = __expf(h0 - m);
    float e1 = __expf(h1 - m);
    float inv = 1.0f / (e0 + e1);
    float p0 = e0 * inv, p1 = e1 * inv;
    int   label = (p1 > p0) ? 1 : 0;       // argmax, first index wins ties
    float score = fmaxf(p0, p1);

    ws_hscore[i] = score;
    ws_valid[i]  = (label == 0 && score >= HUMAN_CONF) ? 1 : 0;

    float W = (float)(*p_w);
    float H = (float)(*p_h);
    float cx = out_bbox[4 * i + 0];
    float cy = out_bbox[4 * i + 1];
    float w  = out_bbox[4 * i + 2];
    float h  = out_bbox[4 * i + 3];
    float x0 = (cx - 0.5f * w) * W;
    float y0 = (cy - 0.5f * h) * H;
    float x1 = (cx + 0.5f * w) * W;
    float y1 = (cy + 0.5f * h) * H;

    ws_boxes[4 * i + 0] = x0;
    ws_boxes[4 * i + 1] = y0;
    ws_boxes[4 * i + 2] = x1;
    ws_boxes[4 * i + 3] = y1;

    out_boxes[4 * i + 0] = (float)(int)x0;
    out_boxes[4 * i + 1] = (float)(int)y0;
    out_boxes[4 * i + 2] = (float)(int)x1;
    out_boxes[4 * i + 3] = (float)(int)y1;
}

// ---------------------------------------------------------------------------
// Kernel 2: per-image greedy NMS. One 256-thread workgroup per batch image.
// ---------------------------------------------------------------------------
__global__ __launch_bounds__(256) void nms_kernel(
    const float* __restrict__ ws_boxes,
    const float* __restrict__ ws_hscore,
    const int*   __restrict__ ws_valid,
    int*   __restrict__ ws_keep,   // [BQ] int
    float* __restrict__ out_keep)  // [BQ] 0/1 as float
{
    __shared__ float key[QPAD];
    __shared__ int   sidx[QPAD];
    __shared__ float sb[QQ][4];
    __shared__ float sa[QQ];
    __shared__ int   sv[QQ];
    __shared__ int   sk[QQ];
    __shared__ int   flag;

    const int b = blockIdx.x;
    const int t = threadIdx.x;
    const float* boxes = ws_boxes + (size_t)b * QQ * 4;

    for (int q = t; q < QPAD; q += 256) {
        if (q < QQ) {
            key[q]  = ws_valid[b * QQ + q] ? ws_hscore[b * QQ + q] : -1e30f;
            sidx[q] = q;
        } else {
            key[q]  = -1e30f;
            sidx[q] = q;
        }
    }
    __syncthreads();

    // Bitonic sort, descending by key
    for (int k = 2; k <= QPAD; k <<= 1) {
        for (int j = k >> 1; j > 0; j >>= 1) {
            for (int i = t; i < QPAD; i += 256) {
                int ixj = i ^ j;
                if (ixj > i) {
                    bool desc = ((i & k) == 0);
                    float ki = key[i], kj = key[ixj];
                    bool do_swap = desc ? (ki < kj) : (ki > kj);
                    if (do_swap) {
                        key[i] = kj; key[ixj] = ki;
                        int tmp = sidx[i]; sidx[i] = sidx[ixj]; sidx[ixj] = tmp;
                    }
                }
            }
            __syncthreads();
        }
    }

    for (int i = t; i < QQ; i += 256) {
        int o = sidx[i];
        float x0 = boxes[4 * o + 0];
        float y0 = boxes[4 * o + 1];
        float x1 = boxes[4 * o + 2];
        float y1 = boxes[4 * o + 3];
        sb[i][0] = x0; sb[i][1] = y0; sb[i][2] = x1; sb[i][3] = y1;
        sa[i] = (x1 - x0) * (y1 - y0);
        sv[i] = ws_valid[b * QQ + o];
        sk[i] = 0;
    }
    __syncthreads();

    for (int i = 0; i < QQ; ++i) {
        if (t == 0) flag = 0;
        __syncthreads();
        float ix0 = sb[i][0], iy0 = sb[i][1], ix1 = sb[i][2], iy1 = sb[i][3];
        float ia = sa[i];
        int local = 0;
        for (int j = t; j < i; j += 256) {
            if (sk[j]) {
                float lx = fmaxf(ix0, sb[j][0]);
                float ly = fmaxf(iy0, sb[j][1]);
                float rx = fminf(ix1, sb[j][2]);
                float ry = fminf(iy1, sb[j][3]);
                float iw = fmaxf(rx - lx, 0.0f);
                float ih = fmaxf(ry - ly, 0.0f);
                float inter = iw * ih;
                float uni = fmaxf(ia + sa[j] - inter, 1e-9f);
                if (inter / uni > IOU_THRESH) local = 1;
            }
        }
        if (local) flag = 1;  // benign same-value race
        __syncthreads();
        if (t == 0) sk[i] = (sv[i] && !flag) ? 1 : 0;
        __syncthreads();
    }

    for (int i = t; i < QQ; i += 256) {
        int o = sidx[i];
        ws_keep[b * QQ + o]  = sk[i];
        out_keep[b * QQ + o] = sk[i] ? 1.0f : 0.0f;
    }
}

// ---------------------------------------------------------------------------
// Kernel 3: prob = out_logits @ Aaug via V_WMMA_F32_16X16X4_F32 (fp32 path,
// matches reference precision), with fused threshold/remap + keep gating.
// 8 waves/WG, each wave owns one 16-row M tile and TWO 16-col N tiles
// (one A fragment feeds two WMMAs -> halves A HBM traffic).
// B panels (2 x 368x16 f32) staged in LDS via the Tensor Data Mover:
// each wave TDM-loads a disjoint 92-row slice; OOB reads beyond tensor_dim
// return zero, which implements the K/N zero padding in hardware.
// ---------------------------------------------------------------------------
__global__ __launch_bounds__(256) void gemm_score_kernel(
    const float* __restrict__ logits,   // [BQ, 365]
    const float* __restrict__ Aaug,     // [365, 365]
    const int*   __restrict__ ws_keep,  // [BQ]
    float* __restrict__ out_scores)     // [BQ, 365]
{
    __shared__ float Bs[NT_PER_WG][KPAD * 16];   // Bs[p][k*16+n]

    const int wave  = threadIdx.x >> 5;
    const int lane  = threadIdx.x & 31;
    const int Mbase = (blockIdx.x * WAVES_PER_WG + wave) * 16;
    const int nt0   = blockIdx.y * NT_PER_WG;    // first N tile of this block

#ifdef USE_TDM
    {
        // Per-wave slice: panel = wave>>2 (0/1), rows [slice*92, slice*92+92)
        const int panel    = wave >> 2;
        const int slice    = wave & 3;
        const int rowstart = slice * 92;
        int cs = (nt0 + panel) * 16;
        const int colstart = (cs > 352) ? 352 : cs;   // clamp pad tile (writes are guarded)
        const int dim0 = CC - colstart;               // remaining cols (zero-fill past this)
        const int dim1 = CC - rowstart;               // remaining rows

        unsigned lds_addr = (unsigned)(uintptr_t)(&Bs[0][0])
                          + (unsigned)((panel * (KPAD * 16) + rowstart * 16) * 4);
        unsigned long long ga = (unsigned long long)(uintptr_t)Aaug
                              + ((unsigned long long)((size_t)rowstart * CC + colstart)) * 4ull;

        u32x4 g0;
        g0[0] = 1u;                                   // count=1, is_restore=0, gather=0
        g0[1] = lds_addr;                             // LDS byte address
        g0[2] = (unsigned)(ga & 0xFFFFFFFFu);         // global_addr[31:0]
        g0[3] = (unsigned)((ga >> 32) & 0x1FFFFFFu)   // global_addr[56:32]
              | (2u << 30);                           // type=2 ("image")

        i32x8 g1;
        g1[0] = (int)(2u << 16);                      // data_size=2 (4 bytes)
        g1[1] = (int)(((unsigned)dim0 & 0xFFFFu) << 16);            // tensor_dim0[15:0]
        g1[2] = (int)((((unsigned)dim0 >> 16) & 0xFFFFu)
              | (((unsigned)dim1 & 0xFFFFu) << 16));                // dim0[31:16] | dim1[15:0]
        g1[3] = (int)((((unsigned)dim1 >> 16) & 0xFFFFu)
              | (16u << 16));                                       // dim1[31:16] | tile_dim0=16
        g1[4] = 92;                                   // tile_dim1 (rows per slice)
        g1[5] = CC;                                   // tensor_dim0_stride[31:0]
        g1[6] = 0;
        g1[7] = 0;
        i32x4 gz4 = {0, 0, 0, 0};
        i32x8 gz8 = {0, 0, 0, 0, 0, 0, 0, 0};

        // amdgpu-toolchain (clang-23) 6-arg form:
        // (u32x4 g0, i32x8 g1, i32x4, i32x4, i32x8, i32 cpol)
        __builtin_amdgcn_tensor_load_to_lds(g0, g1, gz4, gz4, gz8, 0);
        __builtin_amdgcn_s_wait_tensorcnt(0);
    }
    __syncthreads();
#else
    // Fallback: cooperative fill with explicit zero padding
    for (int e = threadIdx.x; e < NT_PER_WG * KPAD * 16; e += 256) {
        int p = e / (KPAD * 16);
        int r = e - p * (KPAD * 16);
        int k = r >> 4;
        int n = (r & 15) + (nt0 + p) * 16;
        Bs[p][r] = (k < CC && n < CC) ? Aaug[k * CC + n] : 0.0f;
    }
    __syncthreads();
#endif

    // A fragment addressing per ISA 32-bit 16x4 layout:
    //   lanes 0-15: M=lane,    VGPR0=K0, VGPR1=K1
    //   lanes 16-31: M=lane-16, VGPR0=K2, VGPR1=K3
    const int mrow  = Mbase + (lane & 15);
    const int khalf = (lane >> 4) * 2;            // 0 or 2
    const int nn    = lane & 15;
    const float* arow = logits + (size_t)mrow * CC;

    v8f c0 = {};
    v8f c1 = {};

    // Main loop: k = 0..363, no guards (k0+khalf+1 <= 363 < 365)
    for (int k0 = 0; k0 < 364; k0 += 4) {
        int ka = k0 + khalf;
        v2f a, b0, b1;
        a.x  = arow[ka];
        a.y  = arow[ka + 1];
        b0.x = Bs[0][ka * 16 + nn];
        b0.y = Bs[0][(ka + 1) * 16 + nn];
        b1.x = Bs[1][ka * 16 + nn];
        b1.y = Bs[1][(ka + 1) * 16 + nn];
        c0 = __builtin_amdgcn_wmma_f32_16x16x4_f32(false, a, false, b0, (short)0, c0, false, false);
        c1 = __builtin_amdgcn_wmma_f32_16x16x4_f32(false, a, false, b1, (short)0, c1, false, false);
    }
    // Tail: k0 = 364 covers k=364 (valid) .. 367 (zero-padded)
    {
        int ka = 364 + khalf;
        v2f a, b0, b1;
        a.x  = (khalf == 0) ? arow[364] : 0.0f;
        a.y  = 0.0f;
        b0.x = Bs[0][ka * 16 + nn];
        b0.y = Bs[0][(ka + 1) * 16 + nn];
        b1.x = Bs[1][ka * 16 + nn];
        b1.y = Bs[1][(ka + 1) * 16 + nn];
        c0 = __builtin_amdgcn_wmma_f32_16x16x4_f32(false, a, false, b0, (short)0, c0, false, false);
        c1 = __builtin_amdgcn_wmma_f32_16x16x4_f32(false, a, false, b1, (short)0, c1, false, false);
    }

    // D layout: VGPR r -> M = r + (lane<16 ? 0 : 8), N = lane & 15
    const int n0   = nt0 * 16 + nn;
    const int n1   = n0 + 16;
    const int mTop = Mbase + ((lane >> 4) * 8);
    #pragma unroll
    for (int r = 0; r < 8; ++r) {
        int m = mTop + r;
        int kp = ws_keep[m];
        float p0 = c0[r];
        float p1 = c1[r];
        float s0 = (kp && p0 >= THRESH) ? (p0 + 1.0f) * 0.5f : 0.0f;
        float s1 = (kp && p1 >= THRESH) ? (p1 + 1.0f) * 0.5f : 0.0f;
        if (n0 < CC) out_scores[(size_t)m * CC + n0] = s0;
        if (n1 < CC) out_scores[(size_t)m * CC + n1] = s1;
    }
}

// ---------------------------------------------------------------------------
extern "C" void kernel_launch(void* const* d_in, const int* in_sizes, int n_in,
                              void* d_out, int out_size, void* d_ws, size_t ws_size,
                              hipStream_t stream) {
    const float* out_human  = (const float*)d_in[0];   // [B,Q,2]
    const float* out_logits = (const float*)d_in[1];   // [B,Q,C]
    const float* out_bbox   = (const float*)d_in[2];   // [B,Q,4]
    const float* Aaug       = (const float*)d_in[3];   // [C,C]
    const int*   p_h        = (const int*)d_in[4];     // scalar
    const int*   p_w        = (const int*)d_in[5];     // scalar

    // d_out: scores [B,Q,C] | boxes [B,Q,4] | keep [B,Q], as float32
    float* out_scores = (float*)d_out;
    float* out_boxes  = out_scores + (size_t)BQ * CC;
    float* out_keep   = out_boxes + (size_t)BQ * 4;

    // Workspace layout
    char* ws = (char*)d_ws;
    float* ws_boxes  = (float*)ws;                      ws += (size_t)BQ * 4 * sizeof(float);
    float* ws_hscore = (float*)ws;                      ws += (size_t)BQ * sizeof(float);
    int*   ws_valid  = (int*)ws;                        ws += (size_t)BQ * sizeof(int);
    int*   ws_keep   = (int*)ws;

    // 1) pointwise prep
    prep_kernel<<<(BQ + 255) / 256, 256, 0, stream>>>(
        out_human, out_bbox, p_h, p_w, ws_boxes, ws_hscore, ws_valid, out_boxes);

    // 2) per-image NMS (one WG per image)
    nms_kernel<<<BB, 256, 0, stream>>>(
        ws_boxes, ws_hscore, ws_valid, ws_keep, out_keep);

    // 3) WMMA fp32 GEMM + fused threshold / keep gating
    dim3 grid(MTILES / WAVES_PER_WG, NGROUPS);
    gemm_score_kernel<<<grid, 256, 0, stream>>>(
        out_logits, Aaug, ws_keep, out_scores);
}